// Cell_46042049413406
// MI455X (gfx1250) — compile-verified
//
#include <hip/hip_runtime.h>
#include <hip/hip_bf16.h>

// CDNA5 (gfx1250) fused quantized conv1d x2 using V_WMMA_F32_16X16X4_F32.
// Exact fp32 math preserves trax-style fake-quant semantics.
//
// Implicit-GEMM mapping per 16-position tile (N = positions, M = out channels
// padded to 16 with zero A-rows, K = reduction over (tap, channel), k-major):
//   conv1: r = k*4 + i  (12 values -> 3 WMMAs of K=4)
//   conv2: r = k*8 + o  (24 values -> 6 WMMAs of K=4)
// A (16x4 f32): lane%16 = M row; lanes<16 hold K {0,1}, lanes>=16 K {2,3}.
// B (4x16 f32): lane%16 = N col; same K split -> each lane supplies 2 scalars,
//   which sit exactly one LDS pitch apart (-> ds_load_2addr_b32 merge).
// C (16x16 f32): VGPR j, lanes<16 => channel j at position lane%16.

typedef __attribute__((ext_vector_type(2))) float v2f;
typedef __attribute__((ext_vector_type(8))) float v8f;

// fake_quant, bits=8: s=128, round-half-up, clip to [-1, 127/128]
__device__ __forceinline__ float fq(float x) {
    float q = floorf(x * 128.0f + 0.5f);
    q = fminf(fmaxf(q, -128.0f), 127.0f);
    return q * 0.0078125f;
}

#define XPITCH 152   // 146 used + pad (delta fits ds_load_2addr offset range)
#define YPITCH 148   // 144 used + pad
#define WAVES_PER_BLOCK 8

__global__ void __launch_bounds__(256)
fused_qconv_wmma(const float* __restrict__ x,
                 const float* __restrict__ w1, const float* __restrict__ b1,
                 const float* __restrict__ gamma, const float* __restrict__ beta,
                 const float* __restrict__ bn_mean, const float* __restrict__ bn_var,
                 const float* __restrict__ w2, const float* __restrict__ b2,
                 float* __restrict__ out, int NB, int L)
{
    __shared__ float lds_x[WAVES_PER_BLOCK][4 * XPITCH];   // quantized x + halo
    __shared__ float lds_y[WAVES_PER_BLOCK][8 * YPITCH];   // quantized y + halo

    const int tid    = threadIdx.x;
    const int wave   = tid >> 5;
    const int lane   = tid & 31;
    const int lane16 = lane & 15;
    const int hi     = lane >> 4;     // 0: lanes 0-15 (K 0,1), 1: lanes 16-31 (K 2,3)
    const int m      = lane16;        // A-matrix row (M) covered by this lane
    float* xlds = &lds_x[wave][0];
    float* ylds = &lds_y[wave][0];

    // ---------------- fold BN + fake-quant weights (per-lane A constants) ----
    // conv1 A: r = k*4 + i ; chunk c has k = c, i = 2*hi + {0,1}
    v2f a1[3];
    {
        float sf = 0.f;
        if (m < 8) sf = gamma[m] / sqrtf(bn_var[m] + 1e-5f);
        #pragma unroll
        for (int c = 0; c < 3; ++c) {
            const int i0 = 2 * hi;
            float e0 = 0.f, e1 = 0.f;
            if (m < 8) {
                e0 = fq(w1[m * 12 + i0 * 3 + c]       * sf);   // (i0,   k=c)
                e1 = fq(w1[m * 12 + (i0 + 1) * 3 + c] * sf);   // (i0+1, k=c)
            }
            a1[c].x = e0; a1[c].y = e1;
        }
    }
    // conv2 A: r = k*8 + o ; chunk c has k = c>>1, o = 4*(c&1) + 2*hi + {0,1}
    v2f a2[6];
    #pragma unroll
    for (int c = 0; c < 6; ++c) {
        const int k  = c >> 1;
        const int o0 = 4 * (c & 1) + 2 * hi;
        float e0 = 0.f, e1 = 0.f;
        if (m < 2) {
            e0 = fq(w2[m * 24 + o0 * 3 + k]);
            e1 = fq(w2[m * 24 + (o0 + 1) * 3 + k]);
        }
        a2[c].x = e0; a2[c].y = e1;
    }
    // biases (every lane keeps all of them; tiny)
    float bq1[8], bq2[2];
    #pragma unroll
    for (int o = 0; o < 8; ++o) {
        float sf = gamma[o] / sqrtf(bn_var[o] + 1e-5f);
        bq1[o] = fq((b1[o] - bn_mean[o]) * sf + beta[o]);
    }
    bq2[0] = fq(b2[0]);
    bq2[1] = fq(b2[1]);

    // ---------------- contiguous per-wave range of 128-position segments -----
    const int segs      = (L + 127) >> 7;
    const int totalIter = NB * segs;
    const int nWaves    = (int)((gridDim.x * blockDim.x) >> 5);
    const int gwave     = (int)((blockIdx.x * blockDim.x + tid) >> 5);
    const int perWave   = (totalIter + nWaves - 1) / nWaves;
    int it    = gwave * perWave;
    int itEnd = it + perWave;
    if (itEnd > totalIter) itEnd = totalIter;

    int bb  = (it < totalIter) ? (it / segs) : 0;   // one division per wave
    int seg = it - bb * segs;

    for (; it < itEnd; ++it) {
        const int zbase = seg << 7;
        const float* xb = x + (size_t)bb * 4 * L;

        // prefetch next segment's x rows (global_prefetch_b8)
        if (seg + 1 < segs) {
            const float* px = xb + ((size_t)(zbase + 128));
            #pragma unroll
            for (int i = 0; i < 4; ++i)
                __builtin_prefetch(px + (size_t)i * L + lane * 4, 0, 0);
        }

        // ---- stage fq(x) once into zero-padded LDS: pos [zbase-9, zbase+137)
        #pragma unroll
        for (int i = 0; i < 4; ++i) {
            const float* xrow = xb + (size_t)i * L;
            float* dst = xlds + i * XPITCH;
            #pragma unroll
            for (int j0 = 0; j0 < 160; j0 += 32) {
                const int j = j0 + lane;
                if (j < 146) {
                    const int pos = zbase - 9 + j;
                    float v = 0.f;
                    if (pos >= 0 && pos < L) v = fq(xrow[pos]);
                    dst[j] = v;
                }
            }
        }

        // ---- conv1: 9 y-tiles of 16 covering [zbase-8, zbase+136) ----------
        #pragma unroll 1
        for (int t = 0; t < 9; ++t) {
            const int p   = zbase - 8 + 16 * t + lane16;  // this lane's position
            const int idx = 16 * t + lane16;              // = pos + k - (zbase-9), k=0
            v8f acc = {};
            #pragma unroll
            for (int c = 0; c < 3; ++c) {                 // k = c
                const int i0 = 2 * hi;
                const int a  = i0 * XPITCH + idx + c;
                v2f b;
                b.x = xlds[a];                            // (i0,   k=c)
                b.y = xlds[a + XPITCH];                   // (i0+1, k=c) -> read2
                acc = __builtin_amdgcn_wmma_f32_16x16x4_f32(
                          false, a1[c], false, b, (short)0, acc, false, false);
            }
            // lanes 0-15 hold channels 0..7 (VGPR j = channel j) at position p
            const bool inb = (p >= 0) && (p < L);
            if (hi == 0) {
                #pragma unroll
                for (int o = 0; o < 8; ++o) {
                    float yv = inb ? fq(acc[o] + bq1[o]) : 0.f;  // conv2 zero-pad
                    ylds[o * YPITCH + idx] = yv;
                }
            }
        }

        // ---- conv2: 8 z-tiles of 16, B from LDS y halo ----------------------
        #pragma unroll 1
        for (int zt = 0; zt < 8; ++zt) {
            const int p    = zbase + 16 * zt + lane16;
            const int base = 16 * zt + lane16 + 7;   // idx of yq[*, p-1] in ylds
            v8f acc = {};
            #pragma unroll
            for (int c = 0; c < 6; ++c) {
                const int k  = c >> 1;
                const int o0 = 4 * (c & 1) + 2 * hi;
                const int a  = o0 * YPITCH + base + k;
                v2f b;
                b.x = ylds[a];                            // (o0,   k)
                b.y = ylds[a + YPITCH];                   // (o0+1, k) -> read2
                acc = __builtin_amdgcn_wmma_f32_16x16x4_f32(
                          false, a2[c], false, b, (short)0, acc, false, false);
            }
            if (hi == 0 && p < L) {
                float z0 = fq(acc[0] + bq2[0]);
                float z1 = fq(acc[1] + bq2[1]);
                out[((size_t)bb * 2 + 0) * L + p] = z0;
                out[((size_t)bb * 2 + 1) * L + p] = z1;
            }
        }

        // incremental (bb, seg) update -- no per-iteration division
        if (++seg == segs) { seg = 0; ++bb; }
    }
}

extern "C" void kernel_launch(void* const* d_in, const int* in_sizes, int n_in,
                              void* d_out, int out_size, void* d_ws, size_t ws_size,
                              hipStream_t stream) {
    const float* x       = (const float*)d_in[0];
    const float* w1      = (const float*)d_in[1];
    const float* b1      = (const float*)d_in[2];
    const float* gamma   = (const float*)d_in[3];
    const float* beta    = (const float*)d_in[4];
    const float* bn_mean = (const float*)d_in[5];
    const float* bn_var  = (const float*)d_in[6];
    const float* w2      = (const float*)d_in[7];
    const float* b2      = (const float*)d_in[8];
    float* out = (float*)d_out;

    const int NB = 16;
    const int L  = in_sizes[0] / (NB * 4);          // 524288 for reference shapes
    const int segs = (L + 127) >> 7;
    const int totalIter = NB * segs;                 // 65536

    int blocks = 2048;                               // 16384 waves, 4 segments each
    if (blocks * WAVES_PER_BLOCK > totalIter)
        blocks = (totalIter + WAVES_PER_BLOCK - 1) / WAVES_PER_BLOCK;
    if (blocks < 1) blocks = 1;

    fused_qconv_wmma<<<blocks, 256, 0, stream>>>(
        x, w1, b1, gamma, beta, bn_mean, bn_var, w2, b2, out, NB, L);
}